// MeshConv_2_86474871537963
// MI455X (gfx1250) — compile-verified
//
#include <hip/hip_runtime.h>
#include <hip/hip_bf16.h>

// Shapes from the reference
#define B_   8
#define F_   16384
#define C_   128
#define OC_  128
#define KN_  6
#define KF_  512      // 4 * C_ flattened GEMM-K
#define TN_  16       // positions per workgroup tile
#define KPR_ 34       // LDS row stride (32 payload + 2 pad floats) per K-pair row

typedef float v2f __attribute__((ext_vector_type(2)));
typedef float v8f __attribute__((ext_vector_type(8)));

// Pre-transpose weight (OC,C,1,4) -> Wt[o][kflat], kflat = ft*128 + c.
// Makes each lane's WMMA A-pair a single contiguous 8-byte load.
__global__ __launch_bounds__(256) void transpose_weight_kernel(
    const float* __restrict__ w, float* __restrict__ wt)
{
    const int idx = blockIdx.x * 256 + threadIdx.x;   // 0..65535
    const int o   = idx >> 9;
    const int r   = idx & 511;                        // kflat
    const int ft  = r >> 7;
    const int c   = r & 127;
    wt[idx] = w[o * 512 + c * 4 + ft];
}

template <bool XW>
__global__ __launch_bounds__(128) void meshconv_wmma_kernel(
    const float* __restrict__ feat,    // (B,F,C)
    const int*   __restrict__ ring,    // (B,F,6)
    const float* __restrict__ wt,      // transposed weight [OC][KF_] (if XW)
    const float* __restrict__ weight,  // original (OC,C,1,4) (if !XW)
    const float* __restrict__ bias,    // (OC)
    float*       __restrict__ out)     // (B,F,OC)
{
    // fcv panel, K-pair interleaved: element (k, n) at [(k>>1)*KPR_ + n*2 + (k&1)]
    __shared__ float fcv2[(KF_ / 2) * KPR_];      // 34,816 B
    __shared__ float stage[4][TN_ * 16];          // per-wave 16x16 staging, 4 KB

    const int tile = blockIdx.x;                  // 0 .. (B*F/16 - 1)
    const int p0   = tile * TN_;                  // base flat position (b*F + f)
    const int tid  = threadIdx.x;                 // 0..127

    // ---------------- Phase 1: gather + derived features -> LDS ----------------
    {
        const int n  = tid >> 3;                  // position in tile, 0..15
        const int cs = (tid & 7) * 16;            // channel base, 16 channels/thread
        const int p  = p0 + n;
        const int b  = p >> 14;                   // p / F_

        int idx[KN_];
        #pragma unroll
        for (int j = 0; j < KN_; ++j) idx[j] = ring[(size_t)p * KN_ + j];

        const float* fb = feat + (size_t)b * F_ * C_;

        #pragma unroll
        for (int cc = 0; cc < 4; ++cc) {
            const int c0 = cs + cc * 4;
            float4 nbv[KN_];
            #pragma unroll
            for (int j = 0; j < KN_; ++j)
                nbv[j] = *(const float4*)(fb + (size_t)idx[j] * C_ + c0);

            const float* nb = reinterpret_cast<const float*>(&nbv[0]);
            float fv[4][4];                       // [stacked feature][q]
            #pragma unroll
            for (int q = 0; q < 4; ++q) {
                const float n0 = nb[0 * 4 + q];
                float s = 0.f;
                #pragma unroll
                for (int j = 1; j < KN_; ++j) s += nb[j * 4 + q];
                float a3 = 0.f;
                #pragma unroll
                for (int i = 1; i < KN_ - 1; ++i)
                    #pragma unroll
                    for (int j = i + 1; j < KN_; ++j)
                        a3 += fabsf(nb[i * 4 + q] - nb[j * 4 + q]);
                fv[0][q] = n0;                                    // fc0
                fv[1][q] = s;                                     // fc1
                fv[2][q] = fabsf((float)(KN_ - 1) * n0 - s);      // fc2
                fv[3][q] = a3;                                    // fc3
            }
            // ds_store_b64 per (feature, channel-pair)
            #pragma unroll
            for (int ft = 0; ft < 4; ++ft)
                #pragma unroll
                for (int qp = 0; qp < 2; ++qp) {
                    const int kp = ft * 64 + (c0 >> 1) + qp;      // K-pair row
                    float2 v = make_float2(fv[ft][qp * 2], fv[ft][qp * 2 + 1]);
                    *(float2*)&fcv2[kp * KPR_ + n * 2] = v;
                }
        }
    }
    __syncthreads();

    // ------- Phase 2: WMMA GEMM, 2 M-tiles per wave sharing one B panel -------
    const int lane = tid & 31;
    const int wave = tid >> 5;                    // 0..3
    const int lg   = lane >> 4;                   // lane group 0/1
    const int ln   = lane & 15;

    const int m0a = wave * 16;                    // M-tile A
    const int m0b = (wave + 4) * 16;              // M-tile B

    v8f acc0 = {0.f, 0.f, 0.f, 0.f, 0.f, 0.f, 0.f, 0.f};
    v8f acc1 = {0.f, 0.f, 0.f, 0.f, 0.f, 0.f, 0.f, 0.f};

    const float* w0t = XW ? (wt + (size_t)(m0a + ln) * KF_) : nullptr;
    const float* w1t = XW ? (wt + (size_t)(m0b + ln) * KF_) : nullptr;
    const float* w0r = XW ? nullptr : (weight + (size_t)(m0a + ln) * 512);
    const float* w1r = XW ? nullptr : (weight + (size_t)(m0b + ln) * 512);

    #pragma unroll 8
    for (int k0 = 0; k0 < KF_; k0 += 4) {
        const int koff = k0 + lg * 2;             // this lane's K pair base (even)

        // B 4x16 operand: one ds_load_b64 (row kp = koff>>1, col ln)
        v2f bm = *(const v2f*)&fcv2[((k0 >> 1) + lg) * KPR_ + ln * 2];

        v2f a0, a1;                               // A 16x4 fp32 layout (ISA 7.12.2)
        if constexpr (XW) {
            a0 = *(const v2f*)(w0t + koff);       // single global_load_b64
            a1 = *(const v2f*)(w1t + koff);
        } else {
            const int ft = koff >> 7;
            const int c  = koff & 127;
            a0.x = w0r[(c + 0) * 4 + ft];  a0.y = w0r[(c + 1) * 4 + ft];
            a1.x = w1r[(c + 0) * 4 + ft];  a1.y = w1r[(c + 1) * 4 + ft];
        }

        acc0 = __builtin_amdgcn_wmma_f32_16x16x4_f32(
            false, a0, false, bm, (short)0, acc0, false, false);
        acc1 = __builtin_amdgcn_wmma_f32_16x16x4_f32(
            false, a1, false, bm, (short)0, acc1, false, false);
    }

    // ---------------- Epilogue: bias, LDS transpose, coalesced stores ---------
    #pragma unroll
    for (int t = 0; t < 2; ++t) {
        const v8f acc = t ? acc1 : acc0;
        const int m0  = t ? m0b : m0a;

        float* st = &stage[wave][0];
        #pragma unroll
        for (int r = 0; r < 8; ++r) {
            const int m = r + lg * 8;             // local row (output channel)
            st[ln * 16 + m] = acc[r] + bias[m0 + m];
        }
        __syncthreads();

        #pragma unroll
        for (int it = 0; it < 2; ++it) {
            const int w2 = it * 32 + lane;        // 0..63 float4 chunks
            const int n  = w2 >> 2;               // position 0..15
            const int ch = (w2 & 3) * 4;          // channel quad within tile
            float4 v = *(float4*)&st[n * 16 + ch];
            *(float4*)(out + (size_t)(p0 + n) * OC_ + m0 + ch) = v;
        }
        __syncthreads();
    }
}

__global__ __launch_bounds__(128) void zero_faces_kernel(
    const int* __restrict__ zm, const int* __restrict__ zf,
    float* __restrict__ out)
{
    const int i = blockIdx.x;                     // 0..1023
    const int b = zm[i];
    const int f = zf[i];
    out[((size_t)b * F_ + f) * OC_ + threadIdx.x] = 0.f;
}

extern "C" void kernel_launch(void* const* d_in, const int* in_sizes, int n_in,
                              void* d_out, int out_size, void* d_ws, size_t ws_size,
                              hipStream_t stream) {
    const float* feat   = (const float*)d_in[0];  // (B,F,C)
    const int*   ring   = (const int*)  d_in[1];  // (B,F,6)
    const float* weight = (const float*)d_in[2];  // (OC,C,1,4)
    const float* bias   = (const float*)d_in[3];  // (OC)
    const int*   zm     = (const int*)  d_in[4];  // (1024)
    const int*   zf     = (const int*)  d_in[5];  // (1024)
    float* out = (float*)d_out;

    const int ntiles = (B_ * F_) / TN_;           // 8192
    const size_t wt_bytes = (size_t)OC_ * KF_ * sizeof(float);  // 256 KB

    if (ws_size >= wt_bytes) {
        float* wt = (float*)d_ws;
        transpose_weight_kernel<<<(OC_ * KF_) / 256, 256, 0, stream>>>(weight, wt);
        meshconv_wmma_kernel<true><<<ntiles, 128, 0, stream>>>(
            feat, ring, wt, weight, bias, out);
    } else {
        meshconv_wmma_kernel<false><<<ntiles, 128, 0, stream>>>(
            feat, ring, nullptr, weight, bias, out);
    }
    zero_faces_kernel<<<1024, 128, 0, stream>>>(zm, zf, out);
    (void)in_sizes; (void)n_in; (void)out_size;
}